// RBFImageCollection_64020782514766
// MI455X (gfx1250) — compile-verified
//
#include <hip/hip_runtime.h>

// CDNA5 / gfx1250: wave32, WMMA (not MFMA).
typedef float v2f __attribute__((ext_vector_type(2)));
typedef float v8f __attribute__((ext_vector_type(8)));

#define NUM_IMAGES 16
#define NUM_RBF    128
#define IMG_ROWS   256
#define IMG_COLS   256
#define PIX        (IMG_ROWS * IMG_COLS)

#define WAVES_PER_BLOCK 8
#define PTILES_PER_WAVE 4
#define PIX_PER_WAVE    (PTILES_PER_WAVE * 16)           // 64
#define PIX_PER_BLOCK   (WAVES_PER_BLOCK * PIX_PER_WAVE) // 512

#define LOG2E 1.4426950408889634f

// exponent2[k,p] = log2e * ( a_k*(x^2+y^2) + b_k*x + c_k*y + d_k )
//   -> one 16x16x4 f32 WMMA per (16-rbf x 16-pixel) tile; log2e folded into
//      the A coefficients so D feeds a raw v_exp_f32 with no pre-mul/fixup.
// out[n,p] = sigmoid( sum_k amp_k * exp2(exponent2[k,p]) )
__global__ __launch_bounds__(256) void RBFImageCollection_64020782514766_kernel(
    const float* __restrict__ centers,      // (16,128,2)
    const float* __restrict__ covariances,  // (16,128,3)  only comp 0 used
    const float* __restrict__ amplitudes,   // (16,128,1)
    float* __restrict__ out)                // (16,256,256) f32
{
    const int n    = blockIdx.y;
    const int lane = (int)(threadIdx.x & 31u);
    const int wave = (int)(threadIdx.x >> 5u);
    const bool hi  = lane >= 16;           // second half-wave
    const int col  = lane & 15;

    const int pixBase = blockIdx.x * PIX_PER_BLOCK + wave * PIX_PER_WAVE;

    // --- B fragments: pixel features for 4 tiles of 16 pixels ---------------
    // B (4x16 f32) layout: vgpr0: lanes0-15 K=0 row (x^2+y^2), lanes16-31 K=2 row (y)
    //                      vgpr1: lanes0-15 K=1 row (x),       lanes16-31 K=3 row (1)
    v2f bfrag[PTILES_PER_WAVE];
#pragma unroll
    for (int t = 0; t < PTILES_PER_WAVE; ++t) {
        int p = pixBase + t * 16 + col;
        float xf = (float)(p & (IMG_COLS - 1));   // x = column
        float yf = (float)(p >> 8);               // y = row
        v2f b;
        b.x = hi ? yf   : (xf * xf + yf * yf);
        b.y = hi ? 1.0f : xf;
        bfrag[t] = b;
    }

    const float* ctr = centers     + n * NUM_RBF * 2;
    const float* cov = covariances + n * NUM_RBF * 3;
    const float* amp = amplitudes  + n * NUM_RBF;

    float acc[PTILES_PER_WAVE] = {0.f, 0.f, 0.f, 0.f};

    for (int kt = 0; kt < NUM_RBF / 16; ++kt) {
        // --- A fragment: RBF coefficients (pre-scaled by log2e) -------------
        // A (16x4 f32) layout: lanes0-15: vgpr0=K0 (a), vgpr1=K1 (b)
        //                      lanes16-31: vgpr0=K2 (c), vgpr1=K3 (d)
        int   kk    = kt * 16 + col;
        float cx    = ctr[kk * 2 + 0];
        float cy    = ctr[kk * 2 + 1];
        float s0    = cov[kk * 3 + 0];
        // log2e * 0.5/sigma^2, sigma = exp(s0)  ->  exp(-2*s0) via raw exp2
        float inv2s = (0.5f * LOG2E) * __builtin_amdgcn_exp2f(-2.0f * LOG2E * s0);
        float ca = -inv2s;
        float cb =  2.0f * inv2s * cx;
        float cc =  2.0f * inv2s * cy;
        float cd = -inv2s * (cx * cx + cy * cy);
        v2f afrag;
        afrag.x = hi ? cc : ca;
        afrag.y = hi ? cd : cb;

        // amp weights for this k-tile; D-tile row M=v (lanes0-15) / M=v+8 (lanes16-31)
        float av[8];
#pragma unroll
        for (int v = 0; v < 8; ++v) {
            float aLo = amp[kt * 16 + v];       // wave-uniform -> scalar load
            float aHi = amp[kt * 16 + 8 + v];
            av[v] = hi ? aHi : aLo;
        }

#pragma unroll
        for (int t = 0; t < PTILES_PER_WAVE; ++t) {
            v8f cz = {};   // fresh exponent tile each time
            // D = A x B + 0 : exact f32 rank-4 exponent GEMM (log2 domain)
            v8f d = __builtin_amdgcn_wmma_f32_16x16x4_f32(
                false, afrag, false, bfrag[t], (short)0, cz, false, false);

            float s = 0.f;
#pragma unroll
            for (int v = 0; v < 8; ++v) {
                s += av[v] * __builtin_amdgcn_exp2f(d[v]);  // bare v_exp_f32
            }
            acc[t] += s;
        }
    }

    // fold the two half-wave row-groups, sigmoid, store 16 pixels per tile
    float* o = out + n * PIX;
#pragma unroll
    for (int t = 0; t < PTILES_PER_WAVE; ++t) {
        float total = acc[t] + __shfl_xor(acc[t], 16, 32);
        if (!hi) {
            float e  = __builtin_amdgcn_exp2f(-LOG2E * total);   // exp(-total)
            float sg = __builtin_amdgcn_rcpf(1.0f + e);          // v_rcp_f32
            o[pixBase + t * 16 + col] = sg;
        }
    }
}

extern "C" void kernel_launch(void* const* d_in, const int* in_sizes, int n_in,
                              void* d_out, int out_size, void* d_ws, size_t ws_size,
                              hipStream_t stream) {
    (void)in_sizes; (void)n_in; (void)d_ws; (void)ws_size; (void)out_size;
    const float* centers     = (const float*)d_in[0];
    const float* covariances = (const float*)d_in[1];
    const float* amplitudes  = (const float*)d_in[2];
    float* out = (float*)d_out;

    dim3 grid(PIX / PIX_PER_BLOCK, NUM_IMAGES);  // (128, 16)
    dim3 block(256);                              // 8 wave32s
    RBFImageCollection_64020782514766_kernel<<<grid, block, 0, stream>>>(
        centers, covariances, amplitudes, out);
}